// DGCF_39926015983740
// MI455X (gfx1250) — compile-verified
//
#include <hip/hip_runtime.h>
#include <stdint.h>

typedef __attribute__((ext_vector_type(2))) float v2f;
typedef __attribute__((ext_vector_type(8))) float v8f;

#define B_    64
#define N_    1024
#define K_    16
#define CIN_  64
#define COUT_ 64
#define H_    512
#define KDIM1 65536   // N_*CIN_  (reduction dim of GEMM1)
#define NDYN  65536   // K_*CIN_*COUT_ (output cols of GEMM2)

static __device__ __forceinline__ v8f wmma_f32(v2f a, v2f b, v8f c) {
    // D = A(16x4 f32) * B(4x16 f32) + C(16x16 f32)
    return __builtin_amdgcn_wmma_f32_16x16x4_f32(
        /*neg_a=*/false, a, /*neg_b=*/false, b,
        /*c_mod=*/(short)0, c, /*reuse_a=*/false, /*reuse_b=*/false);
}

static __device__ __forceinline__ v8f vzero8() {
    v8f z;
    #pragma unroll
    for (int i = 0; i < 8; ++i) z[i] = 0.0f;
    return z;
}

// Async copy 16B: global -> LDS (direct, no VGPR round trip; ASYNCcnt tracked)
static __device__ __forceinline__ void async_g2l_b128(uint32_t lds_byte_addr,
                                                      const void* gaddr) {
    asm volatile("global_load_async_to_lds_b128 %0, %1, off"
                 :: "v"(lds_byte_addr), "v"((uint64_t)(uintptr_t)gaddr)
                 : "memory");
}
static __device__ __forceinline__ void wait_async0() {
    asm volatile("s_wait_asynccnt 0x0" ::: "memory");
}

// ---------------------------------------------------------------------------
// Kernel 1: h = relu(x_flat(64,65536) @ W1(65536,512) + b1)
// grid = 32 (H/16 col tiles), block = 256 (8 waves split K)
// ---------------------------------------------------------------------------
__global__ __launch_bounds__(256) void k1_gemm_relu(
    const float* __restrict__ x, const float* __restrict__ W1,
    const float* __restrict__ b1, float* __restrict__ h)
{
    __shared__ float red[8][4][8][32];   // 32 KB: wave x mtile x vgpr x lane

    const int ntile = blockIdx.x;        // 0..31
    const int tid   = threadIdx.x;
    const int wave  = tid >> 5;          // 0..7
    const int lane  = tid & 31;
    const int M     = lane & 15;
    const int kh    = lane >> 4;

    v8f acc[4];
    #pragma unroll
    for (int m = 0; m < 4; ++m) acc[m] = vzero8();

    const int k0  = wave * (KDIM1 / 8);
    const int k1e = k0 + (KDIM1 / 8);
    const float* w1base = W1 + (size_t)ntile * 16 + M;

    for (int kk = k0; kk < k1e; kk += 4) {
        const int kr = kk + kh * 2;
        v2f bf;
        bf.x = w1base[(size_t)kr * H_];
        bf.y = w1base[(size_t)(kr + 1) * H_];
        #pragma unroll
        for (int m = 0; m < 4; ++m) {
            v2f af = *(const v2f*)(x + (size_t)(m * 16 + M) * KDIM1 + kr);
            acc[m] = wmma_f32(af, bf, acc[m]);
        }
    }

    #pragma unroll
    for (int m = 0; m < 4; ++m)
        #pragma unroll
        for (int v = 0; v < 8; ++v)
            red[wave][m][v][lane] = acc[m][v];
    __syncthreads();

    for (int e = tid; e < 1024; e += 256) {
        const int m = e >> 8;
        const int v = (e >> 5) & 7;
        const int l = e & 31;
        float s = 0.0f;
        #pragma unroll
        for (int w = 0; w < 8; ++w) s += red[w][m][v][l];
        const int row = m * 16 + v + 8 * (l >> 4);
        const int col = ntile * 16 + (l & 15);
        s += b1[col];
        h[(size_t)row * H_ + col] = fmaxf(s, 0.0f);
    }
}

// ---------------------------------------------------------------------------
// Kernel 2: dyn = h(64,512) @ W2(512,65536) + b2   (dyn == kernels, contiguous)
// grid = 1024, block = 128 (4 waves, each owns one 16-col tile)
// ---------------------------------------------------------------------------
__global__ __launch_bounds__(128) void k2_gemm_dyn(
    const float* __restrict__ h, const float* __restrict__ W2,
    const float* __restrict__ b2, float* __restrict__ dyn)
{
    const int tid  = threadIdx.x;
    const int wave = tid >> 5;
    const int lane = tid & 31;
    const int M    = lane & 15;
    const int kh   = lane >> 4;
    const int ntile = blockIdx.x * 4 + wave;    // 0..4095

    v8f acc[4];
    #pragma unroll
    for (int m = 0; m < 4; ++m) acc[m] = vzero8();

    const float* w2base = W2 + (size_t)ntile * 16 + M;

    for (int kk = 0; kk < H_; kk += 4) {
        const int kr = kk + kh * 2;
        __builtin_prefetch((const void*)(w2base + (size_t)(kr + 16) * NDYN), 0, 1);
        v2f bf;
        bf.x = w2base[(size_t)kr * NDYN];
        bf.y = w2base[(size_t)(kr + 1) * NDYN];
        #pragma unroll
        for (int m = 0; m < 4; ++m) {
            v2f af = *(const v2f*)(h + (size_t)(m * 16 + M) * H_ + kr);
            acc[m] = wmma_f32(af, bf, acc[m]);
        }
    }

    const int col = ntile * 16 + M;
    const float bb = b2[col];
    #pragma unroll
    for (int m = 0; m < 4; ++m)
        #pragma unroll
        for (int v = 0; v < 8; ++v) {
            const int row = m * 16 + v + 8 * kh;
            dyn[(size_t)row * NDYN + col] = acc[m][v] + bb;
        }
}

// ---------------------------------------------------------------------------
// Kernel 3: out[b,n,o] = sum_{k,i} xz[b, nb[n,k], i] * kernels[b,k,i,o] + bias[o]
// grid = 1024 (64 b x 16 tile groups), block = 128 (4 waves, one n-tile each)
// kernels[b] viewed as (1024, 64) row-major; 64-row chunks double-buffered in
// LDS via GLOBAL_LOAD_ASYNC_TO_LDS_B128 (one chunk == one neighbor k, CIN==64).
// ---------------------------------------------------------------------------
__global__ __launch_bounds__(128) void k3_gather_einsum(
    const float* __restrict__ x, const int* __restrict__ nb,
    const float* __restrict__ kern, const float* __restrict__ bias,
    float* __restrict__ out)
{
    __shared__ float Bs[2][64 * 64];     // 2 x 16 KB chunks of kernels[b]

    const int b    = blockIdx.x >> 4;    // 0..63
    const int tg   = blockIdx.x & 15;
    const int tid  = threadIdx.x;
    const int wave = tid >> 5;
    const int lane = tid & 31;
    const int M    = lane & 15;
    const int kh   = lane >> 4;
    const int ntile = tg * 4 + wave;     // 0..63
    const int nrow  = ntile * 16 + M;

    const float* kb = kern + (size_t)b * NDYN;        // (1024,64)
    const float* xb = x    + (size_t)b * (N_ * CIN_);

    // LDS byte offset of Bs (low 32 bits of LDS->generic addrspacecast)
    const uint32_t lds0 = (uint32_t)(uintptr_t)(void*)&Bs[0][0];

    // Issue one 64x64 chunk (16 KB) as 8 async b128 transfers per thread.
    auto issue_chunk = [&](int r0, int buf) {
        const char* src = (const char*)(kb + (size_t)r0 * COUT_);
        #pragma unroll
        for (int j = 0; j < 8; ++j) {
            const uint32_t slot = (uint32_t)tid + (uint32_t)j * 128u; // float4 idx
            async_g2l_b128(lds0 + (uint32_t)buf * 16384u + slot * 16u,
                           src + (size_t)slot * 16u);
        }
    };

    v8f acc[4];
    #pragma unroll
    for (int o = 0; o < 4; ++o) acc[o] = vzero8();

    issue_chunk(0, 0);

    for (int c = 0; c < K_; ++c) {                 // 16 chunks of 64 rows
        wait_async0();                             // my chunk-c writes landed
        __syncthreads();                           // everyone's landed; prev buf free
        if (c + 1 < K_) issue_chunk((c + 1) * 64, (c + 1) & 1);

        const float* Bcur = Bs[c & 1];
        const int idx = nb[nrow * K_ + c];         // neighbor for this k-slice
        const bool valid = (idx < N_);             // idx == N -> zero pad row
        const float scale = valid ? 1.0f : 0.0f;
        const float* arow = xb + (size_t)(valid ? idx : 0) * CIN_;

        #pragma unroll
        for (int rr = 0; rr < 64; rr += 4) {
            const int r = rr + kh * 2;
            v2f af = *(const v2f*)(arow + r);
            af.x *= scale; af.y *= scale;
            #pragma unroll
            for (int o = 0; o < 4; ++o) {
                v2f bf;
                bf.x = Bcur[r * COUT_ + o * 16 + M];
                bf.y = Bcur[(r + 1) * COUT_ + o * 16 + M];
                acc[o] = wmma_f32(af, bf, acc[o]);
            }
        }
    }

    #pragma unroll
    for (int o = 0; o < 4; ++o)
        #pragma unroll
        for (int v = 0; v < 8; ++v) {
            const int nloc = v + 8 * kh;
            const int oc   = o * 16 + M;
            out[((size_t)b * N_ + ntile * 16 + nloc) * COUT_ + oc] =
                acc[o][v] + bias[oc];
        }
}

// ---------------------------------------------------------------------------
extern "C" void kernel_launch(void* const* d_in, const int* in_sizes, int n_in,
                              void* d_out, int out_size, void* d_ws, size_t ws_size,
                              hipStream_t stream) {
    const float* x    = (const float*)d_in[0];
    const int*   nb   = (const int*)  d_in[1];
    const float* W1   = (const float*)d_in[2];
    const float* b1   = (const float*)d_in[3];
    const float* W2   = (const float*)d_in[4];
    const float* b2   = (const float*)d_in[5];
    const float* bias = (const float*)d_in[6];
    float* out = (float*)d_out;

    float* h   = (float*)d_ws;           // 64*512 floats = 128 KB
    float* dyn = h + B_ * H_;            // 64*65536 floats = 16.7 MB

    k1_gemm_relu    <<<32,   256, 0, stream>>>(x, W1, b1, h);
    k2_gemm_dyn     <<<1024, 128, 0, stream>>>(h, W2, b2, dyn);
    k3_gather_einsum<<<1024, 128, 0, stream>>>(x, nb, dyn, bias, out);
}